// ChamferLoss_51780125721258
// MI455X (gfx1250) — compile-verified
//
#include <hip/hip_runtime.h>

typedef __attribute__((ext_vector_type(2))) float v2f;
typedef __attribute__((ext_vector_type(8))) float v8f;

#define B_   8
#define N_   4096
#define M_   4096
#define MT_  256            // M / 16 tiles
#define NCHUNKS_ 8          // chunks along N
#define TILES_PER_CHUNK_ 32 // 512 columns / 16
#define PINF_ 0x7F800000u

// ---------------------------------------------------------------------------
// Init: min arrays to +inf bit pattern
// ---------------------------------------------------------------------------
__global__ __launch_bounds__(256)
void chamfer_init(unsigned int* __restrict__ min1, unsigned int* __restrict__ min2) {
    int i = blockIdx.x * 256 + threadIdx.x;
    if (i < B_ * N_) { min1[i] = PINF_; min2[i] = PINF_; }
}

// ---------------------------------------------------------------------------
// Main: one wave per (b, m-tile, n-chunk). WMMA computes the full squared
// distance tile directly: D = C + (-2*y) * x with C = ysq[m] + xsq[n].
// min(sqrt(eps+max(d,0))) is computed as sqrt(eps+max(min(d),0)) at the end.
// ---------------------------------------------------------------------------
__global__ __launch_bounds__(256)
void chamfer_tiles(const float* __restrict__ x, const float* __restrict__ y,
                   unsigned int* __restrict__ min1, unsigned int* __restrict__ min2) {
    const int lane  = threadIdx.x & 31;
    const int w     = blockIdx.x * 8 + (threadIdx.x >> 5);   // global wave id
    const int b     = w >> 11;                               // / (MT_ * NCHUNKS_)
    const int rem   = w & 2047;
    const int mt    = rem >> 3;
    const int chunk = rem & 7;
    const int m0    = mt << 4;
    const int nbase = chunk * (TILES_PER_CHUNK_ * 16);

    const int  row = lane & 15;
    const bool hi  = (lane >= 16);

    // ---- A matrix: 16x4 f32, rows = y tile, fold in the -2 factor.
    // ISA layout: VGPR0 = K0 (lanes 0-15) / K2 (lanes 16-31); VGPR1 = K1 / K3(pad=0)
    const float* yr = y + (size_t)(b * M_ + m0 + row) * 3;
    const float y0 = yr[0], y1c = yr[1], y2 = yr[2];
    const float ysq_own = y0 * y0 + y1c * y1c + y2 * y2;
    v2f A;
    A.x = -2.0f * (hi ? y2 : y0);
    A.y = hi ? 0.0f : (-2.0f * y1c);

    // Each lane needs ysq for rows (v + 8*hi), v=0..7 (C-matrix layout).
    float ysq[8];
#pragma unroll
    for (int v = 0; v < 8; ++v)
        ysq[v] = __shfl(ysq_own, v + (hi ? 8 : 0), 32);

    float rowmin[8];
#pragma unroll
    for (int v = 0; v < 8; ++v) rowmin[v] = __uint_as_float(PINF_);

    // ---- software-pipelined x loads: preload tile 0
    const float* x_b = x + (size_t)b * N_ * 3;
    {
        const float* xr0 = x_b + (size_t)(nbase + row) * 3;
        float cx0 = xr0[0], cx1 = xr0[1], cx2 = xr0[2];

        for (int t = 0; t < TILES_PER_CHUNK_; ++t) {
            // issue next tile's load early (wraps on last iter; harmless)
            const int tn = (t + 1) & (TILES_PER_CHUNK_ - 1);
            const float* xrn = x_b + (size_t)(nbase + tn * 16 + row) * 3;
            const float nx0 = xrn[0], nx1 = xrn[1], nx2 = xrn[2];

            // ---- B matrix: 4x16 f32, columns = x tile (mirrors A layout).
            const float xsq = cx0 * cx0 + cx1 * cx1 + cx2 * cx2;
            v2f Bm;
            Bm.x = hi ? cx2 : cx0;
            Bm.y = hi ? 0.0f : cx1;

            // ---- C init: ysq[m] + xsq[n]
            v8f c;
#pragma unroll
            for (int v = 0; v < 8; ++v) c[v] = ysq[v] + xsq;

            // D = A*B + C  ->  squared distances (unclamped)
            v8f d = __builtin_amdgcn_wmma_f32_16x16x4_f32(
                /*neg_a=*/false, A, /*neg_b=*/false, Bm,
                /*c_mod=*/(short)0, c, /*reuse_a=*/false, /*reuse_b=*/false);

            // mins over the 8 rows this lane holds (clamp hoisted out)
            float colmin = d[0];
#pragma unroll
            for (int v = 1; v < 8; ++v) colmin = fminf(colmin, d[v]);
#pragma unroll
            for (int v = 0; v < 8; ++v) rowmin[v] = fminf(rowmin[v], d[v]);

            // all 32 lanes atomic-min: lanes l and l+16 share a column address,
            // hardware combines the two half-tile mins. d2 >= 0 so uint order == float order.
            colmin = fmaxf(colmin, 0.0f);
            atomicMin(min1 + (size_t)b * N_ + nbase + t * 16 + row,
                      __float_as_uint(colmin));

            cx0 = nx0; cx1 = nx1; cx2 = nx2;
        }
    }

    // ---- row mins: butterfly reduce across the 16 columns inside each half
#pragma unroll
    for (int v = 0; v < 8; ++v) {
        float r = rowmin[v];
        r = fminf(r, __shfl_xor(r, 1, 32));
        r = fminf(r, __shfl_xor(r, 2, 32));
        r = fminf(r, __shfl_xor(r, 4, 32));
        r = fminf(r, __shfl_xor(r, 8, 32));
        rowmin[v] = fmaxf(r, 0.0f);
    }
    if (row == 0) {
        const int mb = m0 + (hi ? 8 : 0);
#pragma unroll
        for (int v = 0; v < 8; ++v)
            atomicMin(min2 + (size_t)b * M_ + mb + v, __float_as_uint(rowmin[v]));
    }
}

// ---------------------------------------------------------------------------
// Finish: sqrt(eps + min_d2), mean over both directions (single block).
// ---------------------------------------------------------------------------
__global__ __launch_bounds__(256)
void chamfer_finish(const unsigned int* __restrict__ min1,
                    const unsigned int* __restrict__ min2,
                    float* __restrict__ out) {
    __shared__ float sdata[256];
    const int tid = threadIdx.x;
    float s = 0.0f;
    for (int i = tid; i < B_ * N_; i += 256)
        s += sqrtf(1e-8f + __uint_as_float(min1[i]));
    for (int i = tid; i < B_ * M_; i += 256)
        s += sqrtf(1e-8f + __uint_as_float(min2[i]));
    sdata[tid] = s;
    __syncthreads();
    for (int off = 128; off > 0; off >>= 1) {
        if (tid < off) sdata[tid] += sdata[tid + off];
        __syncthreads();
    }
    if (tid == 0) out[0] = sdata[0] / (float)(B_ * N_);
}

// ---------------------------------------------------------------------------
extern "C" void kernel_launch(void* const* d_in, const int* in_sizes, int n_in,
                              void* d_out, int out_size, void* d_ws, size_t ws_size,
                              hipStream_t stream) {
    const float* x = (const float*)d_in[0];   // x1: (B, N, 3)
    const float* y = (const float*)d_in[1];   // y1: (B, M, 3)
    unsigned int* min1 = (unsigned int*)d_ws;           // B*N floats
    unsigned int* min2 = min1 + (size_t)B_ * N_;        // B*M floats
    float* out = (float*)d_out;

    chamfer_init<<<(B_ * N_ + 255) / 256, 256, 0, stream>>>(min1, min2);
    chamfer_tiles<<<(B_ * MT_ * NCHUNKS_) / 8, 256, 0, stream>>>(x, y, min1, min2);
    chamfer_finish<<<1, 256, 0, stream>>>(min1, min2, out);
}